// HTMLBasedPooling_86904368267404
// MI455X (gfx1250) — compile-verified
//
#include <hip/hip_runtime.h>

// Problem constants (from reference): B=8, NUM_TAGS=1024, SEQ_LEN=4096, HIDDEN=256
#define BB 8
#define TT 1024
#define SS 4096
#define HH 256

typedef float v2f __attribute__((ext_vector_type(2)));
typedef float v8f __attribute__((ext_vector_type(8)));

// ---------------------------------------------------------------------------
// Kernel 0: initialize workspace (harness poisons d_ws with 0xAA)
// ---------------------------------------------------------------------------
__global__ void k_init(int* __restrict__ last, int* __restrict__ cnt) {
    int i = blockIdx.x * blockDim.x + threadIdx.x;
    if (i < BB * SS) last[i] = -1;
    if (i < BB * TT) cnt[i] = 0;
}

// ---------------------------------------------------------------------------
// Kernel 1: last[b,s] = max { j : tag_to_token[b,j,s] > 0 }  (else -1)
// One streaming pass over 128MB with b128 loads; tag chunks of 128 for
// parallelism (256 blocks), merged with atomicMax. Fully coalesced.
// ---------------------------------------------------------------------------
__global__ void __launch_bounds__(256) k_scan(const float* __restrict__ tag,
                                              int* __restrict__ last) {
    const int q  = blockIdx.x * 256 + threadIdx.x;   // quad index: 0..SS/4-1
    const int j0 = blockIdx.y * 128;                 // tag chunk base
    const int b  = blockIdx.z;
    const float4* base = (const float4*)(tag + ((size_t)b * TT + j0) * SS) + q;
    int b0 = -1, b1 = -1, b2 = -1, b3 = -1;
#pragma unroll 4
    for (int j = 0; j < 128; ++j) {
        float4 v = base[(size_t)j * (SS / 4)];
        const int jj = j0 + j;                        // ascending -> max
        if (v.x > 0.0f) b0 = jj;
        if (v.y > 0.0f) b1 = jj;
        if (v.z > 0.0f) b2 = jj;
        if (v.w > 0.0f) b3 = jj;
    }
    int* dst = last + b * SS + q * 4;
    if (b0 >= 0) atomicMax(dst + 0, b0);
    if (b1 >= 0) atomicMax(dst + 1, b1);
    if (b2 >= 0) atomicMax(dst + 2, b2);
    if (b3 >= 0) atomicMax(dst + 3, b3);
}

// ---------------------------------------------------------------------------
// Kernel 2: cnt[b,t] = #{ s : last[b,s] == t }   (row sums of temp)
// ---------------------------------------------------------------------------
__global__ void k_count(const int* __restrict__ last, int* __restrict__ cnt) {
    int i = blockIdx.x * blockDim.x + threadIdx.x;
    if (i >= BB * SS) return;
    int t = last[i];
    if (t >= 0) atomicAdd(&cnt[(i / SS) * TT + t], 1);
}

// ---------------------------------------------------------------------------
// Kernel 3: out[b] = rownorm(onehot(last[b])) x inputs[b] + seq_emb
//
// Per workgroup (8 waves, 32(M) x 256(N) tile): deterministically compact the
// tokens whose last-tag falls in [tileM, tileM+32) into an LDS list of
// (token, tag) pairs, then run the WMMA f32 16x16x4 K-loop over ONLY that
// list (~count/4 chunks instead of SS/4). Exact for any input: omitted
// tokens contribute zero to this tile's rows. Fixed compaction order keeps
// fp accumulation deterministic. Branch-free inner loop -> EXEC all-1s.
// ---------------------------------------------------------------------------
__global__ void __launch_bounds__(256) k_gemm(const float* __restrict__ inputs,
                                              const float* __restrict__ seq_emb,
                                              const int*   __restrict__ last,
                                              const int*   __restrict__ cnt,
                                              float*       __restrict__ out) {
    __shared__ int s_scan[256];
    __shared__ int s_pairs[2 * (SS + 4)];   // (token, tag) pairs, ~32KB worst case

    const int b     = blockIdx.y;
    const int tileM = blockIdx.x * 32;
    const int tid   = threadIdx.x;
    const int* lastB = last + b * SS;

    // ---- pass 1: classify 16 strided tokens per thread (coalesced) --------
    int lv[16];
#pragma unroll
    for (int i = 0; i < 16; ++i) lv[i] = lastB[i * 256 + tid];
    int myCount = 0;
#pragma unroll
    for (int i = 0; i < 16; ++i) myCount += ((unsigned)(lv[i] - tileM) < 32u) ? 1 : 0;

    // ---- block-wide inclusive scan (Hillis-Steele) ------------------------
    s_scan[tid] = myCount;
    __syncthreads();
    for (int off = 1; off < 256; off <<= 1) {
        int v   = s_scan[tid];
        int add = (tid >= off) ? s_scan[tid - off] : 0;
        __syncthreads();
        s_scan[tid] = v + add;
        __syncthreads();
    }
    int pos = s_scan[tid] - myCount;           // exclusive offset
    const int total = s_scan[255];

    // ---- pass 2: ordered compaction into LDS ------------------------------
#pragma unroll
    for (int i = 0; i < 16; ++i) {
        if ((unsigned)(lv[i] - tileM) < 32u) {
            s_pairs[2 * pos + 0] = i * 256 + tid;  // token index s
            s_pairs[2 * pos + 1] = lv[i];          // its tag
            ++pos;
        }
    }
    if (tid < 4) {                                 // sentinel padding
        s_pairs[2 * (total + tid) + 0] = 0;        // safe B address
        s_pairs[2 * (total + tid) + 1] = -1;       // never matches a row
    }
    __syncthreads();

    const int rounds = (total + 3) & ~3;

    // ---- WMMA K-loop over compacted token list ----------------------------
    const int wave = tid >> 5;
    const int lane = tid & 31;
    const int half = lane >> 4;    // K-pair selector per WMMA f32 layout
    const int l15  = lane & 15;    // M for A-frag, N for B/C/D frags

    const int mBase = tileM + (wave >> 2) * 16;  // this wave's 16-row band
    const int n0    = (wave & 3) * 64;           // this wave's 64-col strip
    const int rowA  = mBase + l15;

    const float* inp = inputs + (size_t)b * SS * HH;

    v8f acc0 = {}, acc1 = {}, acc2 = {}, acc3 = {};

    for (int i0 = 0; i0 < rounds; i0 += 4) {
        // this half-wave's two (token, tag) pairs: one b128 LDS broadcast
        const int4 pk = *(const int4*)(&s_pairs[2 * i0 + 4 * half]);
        const int sA0 = pk.x, lA0 = pk.y;
        const int sA1 = pk.z, lA1 = pk.w;

        // A-frag (16x4 f32): lanes 0-15 hold K=i0,i0+1; lanes 16-31 K=i0+2,i0+3
        v2f a;
        a.x = (lA0 == rowA) ? 1.0f : 0.0f;
        a.y = (lA1 == rowA) ? 1.0f : 0.0f;

        // B-frags (4x16 f32): N = l15, VGPR0/1 = gathered rows sA0 / sA1
        const float* r0 = inp + (size_t)sA0 * HH;
        const float* r1 = inp + (size_t)sA1 * HH;

        v2f b0; b0.x = r0[n0 +  0 + l15]; b0.y = r1[n0 +  0 + l15];
        v2f b1; b1.x = r0[n0 + 16 + l15]; b1.y = r1[n0 + 16 + l15];
        v2f b2; b2.x = r0[n0 + 32 + l15]; b2.y = r1[n0 + 32 + l15];
        v2f b3; b3.x = r0[n0 + 48 + l15]; b3.y = r1[n0 + 48 + l15];

        acc0 = __builtin_amdgcn_wmma_f32_16x16x4_f32(false, a, false, b0,
                                                     (short)0, acc0, false, false);
        acc1 = __builtin_amdgcn_wmma_f32_16x16x4_f32(false, a, false, b1,
                                                     (short)0, acc1, false, false);
        acc2 = __builtin_amdgcn_wmma_f32_16x16x4_f32(false, a, false, b2,
                                                     (short)0, acc2, false, false);
        acc3 = __builtin_amdgcn_wmma_f32_16x16x4_f32(false, a, false, b3,
                                                     (short)0, acc3, false, false);
    }

    // ---- epilogue: row = mBase + half*8 + r, col = n0 + j*16 + l15 --------
    float scale[8];
#pragma unroll
    for (int r = 0; r < 8; ++r) {
        const int row = mBase + half * 8 + r;
        const int c   = cnt[b * TT + row];
        scale[r] = 1.0f / (float)(c > 0 ? c : 1);
    }

    const v8f accs[4] = {acc0, acc1, acc2, acc3};
#pragma unroll
    for (int j = 0; j < 4; ++j) {
        const int col = n0 + j * 16 + l15;
#pragma unroll
        for (int r = 0; r < 8; ++r) {
            const int row = mBase + half * 8 + r;
            const float v = accs[j][r] * scale[r] + seq_emb[row * HH + col];
            out[((size_t)b * TT + row) * HH + col] = v;
        }
    }
}

// ---------------------------------------------------------------------------
// Launch
// ---------------------------------------------------------------------------
extern "C" void kernel_launch(void* const* d_in, const int* in_sizes, int n_in,
                              void* d_out, int out_size, void* d_ws, size_t ws_size,
                              hipStream_t stream) {
    (void)in_sizes; (void)n_in; (void)out_size; (void)ws_size;

    const float* inputs  = (const float*)d_in[0];  // [B,S,H] f32
    const float* tag     = (const float*)d_in[1];  // [B,T,S] f32
    const float* seq_emb = (const float*)d_in[2];  // [T,H]   f32
    float*       out     = (float*)d_out;          // [B,T,H] f32

    int* last = (int*)d_ws;         // B*S ints = 128KB
    int* cnt  = last + BB * SS;     // B*T ints = 32KB

    k_init <<<dim3((BB * SS + 255) / 256), 256, 0, stream>>>(last, cnt);
    k_scan <<<dim3(SS / 1024, TT / 128, BB), 256, 0, stream>>>(tag, last);
    k_count<<<dim3((BB * SS + 255) / 256), 256, 0, stream>>>(last, cnt);
    k_gemm <<<dim3(TT / 32, BB),           256, 0, stream>>>(inputs, seq_emb,
                                                             last, cnt, out);
}